// GCN_Simple_53575422050307
// MI455X (gfx1250) — compile-verified
//
#include <hip/hip_runtime.h>
#include <hip/hip_bf16.h>

typedef __attribute__((ext_vector_type(2))) float v2f;
typedef __attribute__((ext_vector_type(8))) float v8f;

#define HID 64

// -------- degree count: deg[dst[e]] += 1 --------
__global__ void k_deg(const int* __restrict__ dst, unsigned* __restrict__ deg, int nE) {
  int e = blockIdx.x * blockDim.x + threadIdx.x;
  if (e < nE) atomicAdd(&deg[dst[e]], 1u);
}

// -------- dinv = rsqrt(deg + 1)  (self-loop adds 1, so deg+1 >= 1) --------
__global__ void k_dinv(const unsigned* __restrict__ deg, float* __restrict__ dinv, int n) {
  int i = blockIdx.x * blockDim.x + threadIdx.x;
  if (i < n) dinv[i] = rsqrtf((float)(deg[i] + 1u));
}

// -------- C[nRows,64] = A[nRows,K] @ B[K,64] using V_WMMA_F32_16X16X4_F32 --------
// One wave computes one 16x16 output tile. nRows % 16 == 0, HID == 64.
__global__ void k_gemm_wmma(const float* __restrict__ A, const float* __restrict__ B,
                            float* __restrict__ C, int nRows, int K) {
  int wave = (int)((blockIdx.x * blockDim.x + threadIdx.x) >> 5);
  int lane = threadIdx.x & 31;
  const int nColTiles = HID / 16;               // 4
  int nRowTiles = nRows / 16;
  if (wave >= nRowTiles * nColTiles) return;
  int row0 = (wave / nColTiles) * 16;
  int col0 = (wave % nColTiles) * 16;
  int half = lane >> 4;                          // 0: lanes 0-15, 1: lanes 16-31
  int lo   = lane & 15;

  // A frag: lane.lo = M row; half selects K pair {2h, 2h+1} within each K=4 step
  const float* arow = A + (size_t)(row0 + lo) * K;
  // B frag: lane.lo = N col; same K-pair selection
  const float* bcol = B + col0 + lo;

  v8f acc = {};
  for (int k0 = 0; k0 < K; k0 += 4) {
    int ka = k0 + 2 * half;
    v2f a, b;
    a.x = arow[ka];
    a.y = arow[ka + 1];
    b.x = bcol[(size_t)ka * HID];
    b.y = bcol[(size_t)(ka + 1) * HID];
    // 8 args: (neg_a, A, neg_b, B, c_mod, C, reuse_a, reuse_b)
    acc = __builtin_amdgcn_wmma_f32_16x16x4_f32(false, a, false, b, (short)0, acc,
                                                false, false);
  }
  // D layout: VGPR v holds row M = v + 8*half, col N = lo
  float* crow = C + (size_t)(row0 + 8 * half) * HID + col0 + lo;
#pragma unroll
  for (int v = 0; v < 8; ++v) crow[(size_t)v * HID] = acc[v];
}

// -------- edge scatter: agg[dst] += xw[src] * dinv[s]*dinv[d], one wave per edge --------
__global__ void k_scatter(const float* __restrict__ xw, const int* __restrict__ src,
                          const int* __restrict__ dst, const float* __restrict__ dinv,
                          float* __restrict__ agg, int nE) {
  int e = (int)((blockIdx.x * blockDim.x + threadIdx.x) >> 5);
  int lane = threadIdx.x & 31;
  if (e >= nE) return;
  int s = src[e], d = dst[e];
  float norm = dinv[s] * dinv[d];
  const float* xs = xw + (size_t)s * HID;
  float* ad = agg + (size_t)d * HID;
  float v0 = xs[lane]      * norm;
  float v1 = xs[lane + 32] * norm;
  atomicAdd(&ad[lane],      v0);
  atomicAdd(&ad[lane + 32], v1);
}

// -------- fused self-loop + bias + ReLU (in place on agg) --------
__global__ void k_sbr(const float* __restrict__ xw, const float* __restrict__ dinv,
                      const float* __restrict__ bias, float* __restrict__ agg, int n) {
  int t = blockIdx.x * blockDim.x + threadIdx.x;   // n*HID threads (6.4M, fits int)
  if (t >= n * HID) return;
  int i = t >> 6, f = t & 63;
  float di = dinv[i];
  float v = agg[t] + xw[t] * di * di + bias[f];
  agg[t] = v > 0.f ? v : 0.f;
}

// -------- global mean pool: sums[batch[i]] += h[i]; cnts[batch[i]] += 1 --------
__global__ void k_pool(const float* __restrict__ h, const int* __restrict__ batch,
                       float* __restrict__ sums, unsigned* __restrict__ cnts, int n) {
  int i = (int)((blockIdx.x * blockDim.x + threadIdx.x) >> 5);
  int lane = threadIdx.x & 31;
  if (i >= n) return;
  int g = batch[i];
  const float* hi = h + (size_t)i * HID;
  atomicAdd(&sums[g * HID + lane],      hi[lane]);
  atomicAdd(&sums[g * HID + lane + 32], hi[lane + 32]);
  if (lane == 0) atomicAdd(&cnts[g], 1u);
}

// -------- out[g,c] = bl[c] + (sums[g,:] / max(cnt,1)) . Wl[:,c] --------
__global__ void k_final(const float* __restrict__ sums, const unsigned* __restrict__ cnts,
                        const float* __restrict__ Wl, const float* __restrict__ bl,
                        float* __restrict__ out, int G, int C) {
  int t = blockIdx.x * blockDim.x + threadIdx.x;
  if (t >= G * C) return;
  int g = t / C, c = t % C;
  float inv = 1.f / fmaxf((float)cnts[g], 1.f);
  float acc = bl[c];
  for (int f = 0; f < HID; ++f) acc += sums[g * HID + f] * inv * Wl[f * C + c];
  out[t] = acc;
}

extern "C" void kernel_launch(void* const* d_in, const int* in_sizes, int n_in,
                              void* d_out, int out_size, void* d_ws, size_t ws_size,
                              hipStream_t stream) {
  (void)n_in; (void)ws_size;
  const float* x          = (const float*)d_in[0];
  const int*   edge_index = (const int*)d_in[1];
  const int*   batch      = (const int*)d_in[2];
  const float* W1         = (const float*)d_in[3];
  const float* b1         = (const float*)d_in[4];
  const float* W2         = (const float*)d_in[5];
  const float* b2         = (const float*)d_in[6];
  const float* Wl         = (const float*)d_in[7];
  const float* bl         = (const float*)d_in[8];
  float* out = (float*)d_out;

  const int FIN  = 128;
  const int N    = in_sizes[0] / FIN;      // 100000 (multiple of 16)
  const int nE   = in_sizes[1] / 2;        // 3200000
  const int nCls = in_sizes[8];            // 10
  const int G    = out_size / nCls;        // 128
  const int* src = edge_index;             // row 0 of [2, E]
  const int* dst = edge_index + nE;        // row 1

  // workspace carve-up (~78 MB)
  float*    f0   = (float*)d_ws;                       // xw buffer   [N, 64]
  float*    f1   = f0 + (size_t)N * HID;               // agg/h layer1 [N, 64]
  float*    f2   = f1 + (size_t)N * HID;               // agg/h layer2 [N, 64]
  float*    dinv = f2 + (size_t)N * HID;               // [N]
  unsigned* deg  = (unsigned*)(dinv + N);              // [N]
  float*    sums = (float*)(deg + N);                  // [G, 64]
  unsigned* cnts = (unsigned*)(sums + (size_t)G * HID);// [G]

  hipMemsetAsync(deg,  0, (size_t)N * sizeof(unsigned),        stream);
  hipMemsetAsync(f1,   0, (size_t)N * HID * sizeof(float),     stream);
  hipMemsetAsync(f2,   0, (size_t)N * HID * sizeof(float),     stream);
  hipMemsetAsync(sums, 0, (size_t)G * HID * sizeof(float),     stream);
  hipMemsetAsync(cnts, 0, (size_t)G * sizeof(unsigned),        stream);

  // degrees + symmetric norm
  k_deg <<<(nE + 255) / 256, 256, 0, stream>>>(dst, deg, nE);
  k_dinv<<<(N + 255) / 256, 256, 0, stream>>>(deg, dinv, N);

  const int tiles   = (N / 16) * (HID / 16);           // waves needed per GEMM
  const int gemmBlk = (tiles * 32 + 255) / 256;
  const int edgeBlk = (int)(((long long)nE * 32 + 255) / 256);
  const int nodeBlk = (int)(((long long)N * 32 + 255) / 256);

  // layer 1
  k_gemm_wmma<<<gemmBlk, 256, 0, stream>>>(x, W1, f0, N, FIN);
  k_scatter  <<<edgeBlk, 256, 0, stream>>>(f0, src, dst, dinv, f1, nE);
  k_sbr      <<<(N * HID + 255) / 256, 256, 0, stream>>>(f0, dinv, b1, f1, N);

  // layer 2
  k_gemm_wmma<<<gemmBlk, 256, 0, stream>>>(f1, W2, f0, N, HID);
  k_scatter  <<<edgeBlk, 256, 0, stream>>>(f0, src, dst, dinv, f2, nE);
  k_sbr      <<<(N * HID + 255) / 256, 256, 0, stream>>>(f0, dinv, b2, f2, N);

  // mean pool + classifier head
  k_pool <<<nodeBlk, 256, 0, stream>>>(f2, batch, sums, cnts, N);
  k_final<<<(G * nCls + 255) / 256, 256, 0, stream>>>(sums, cnts, Wl, bl, out, G, nCls);
}